// VQVAE_52999896432728
// MI455X (gfx1250) — compile-verified
//
#include <hip/hip_runtime.h>

typedef __attribute__((ext_vector_type(2))) float v2f;
typedef __attribute__((ext_vector_type(8))) float v8f;

// Problem constants (B=16, N=1024, D=512, K=8192)
#define MROWS   16384          // B*N
#define DDIM    512
#define KCODES  8192
#define KTILES  (KCODES / 16)  // 512
#define KSPLIT  4              // K slices per M-block
#define KT_PER  (KTILES / KSPLIT)  // 128 k-tiles per slice
#define ZPAD    514            // LDS row stride (2-float pad -> conflict-free b64)

// workspace layout (floats): [0,8192) e2 | partD 16384*4 | partI 16384*4 (as int)
#define WS_E2    0
#define WS_PARTD 8192
#define WS_PARTI (8192 + MROWS * KSPLIT)

// ---------------------------------------------------------------------------
// Kernel 1: e2[k] = sum_d codebook[k][d]^2   (one wave per code row)
// ---------------------------------------------------------------------------
__global__ __launch_bounds__(256) void vq_e2_kernel(const float* __restrict__ cb,
                                                    float* __restrict__ e2) {
  const int lane = threadIdx.x & 31;
  const int wave = threadIdx.x >> 5;
  const int row  = blockIdx.x * 8 + wave;           // 0..8191
  const float4* r = (const float4*)(cb + (size_t)row * DDIM);
  float s = 0.f;
#pragma unroll
  for (int i = 0; i < 4; ++i) {                     // 128 float4 / 32 lanes
    float4 v = r[lane + 32 * i];
    s += v.x * v.x + v.y * v.y + v.z * v.z + v.w * v.w;
  }
#pragma unroll
  for (int off = 16; off > 0; off >>= 1) s += __shfl_xor(s, off, 32);
  if (lane == 0) e2[row] = s;
}

// ---------------------------------------------------------------------------
// Kernel 2: block = 128 rows (8 waves x 16-row subtiles), one K slice.
// All waves sweep the SAME k-tile order -> codebook B reads shared in WGP$.
// Emits partial (bestDist, bestIdx) per row per slice.
// ---------------------------------------------------------------------------
__global__ __launch_bounds__(256) void vq_main_kernel(const float* __restrict__ z_e,
                                                      const float* __restrict__ cb,
                                                      const float* __restrict__ e2,
                                                      float* __restrict__ partD,
                                                      int*   __restrict__ partI) {
  __shared__ float zls[8][16][ZPAD];   // 8 x 16-row z subtiles (~263 KB)
  __shared__ float z2s[8][16];         // per-row ||z||^2

  const int tid    = threadIdx.x;
  const int lane   = tid & 31;
  const int wave   = tid >> 5;
  const int mblock = blockIdx.x;                     // 0..127 (128 rows each)
  const int ks     = blockIdx.y;                     // 0..3   K slice
  const size_t zbase = (size_t)mblock * 128 * DDIM;

  // ---- stage 128x512 z rows into LDS (coalesced float4) ----
#pragma unroll
  for (int i = 0; i < 64; ++i) {
    int f4  = tid + 256 * i;                         // 0..16383
    int row = f4 >> 7;                               // 0..127
    int col = (f4 & 127) << 2;
    float4 v = *(const float4*)(z_e + zbase + (size_t)row * DDIM + col);
    float* d = &zls[row >> 4][row & 15][col];
    d[0] = v.x; d[1] = v.y; d[2] = v.z; d[3] = v.w;
  }
  __syncthreads();

  // ---- per-row ||z||^2 for this wave's 16 rows ----
  for (int r = 0; r < 16; ++r) {
    float s = 0.f;
#pragma unroll
    for (int i = 0; i < 16; ++i) { float v = zls[wave][r][lane + 32 * i]; s += v * v; }
#pragma unroll
    for (int off = 16; off > 0; off >>= 1) s += __shfl_xor(s, off, 32);
    if (lane == 0) z2s[wave][r] = s;
  }
  __syncthreads();

  // ---- main loop: every wave sweeps k-tiles ks*128 .. ks*128+127 in order ----
  const int nsub = lane & 15;                 // B column / A row within tile
  const int hi   = lane >> 4;                 // lane half -> K offset 0 / 2
  const float* arow = &zls[wave][nsub][hi * 2];

  float bestD[8]; int bestI[8];
#pragma unroll
  for (int g = 0; g < 8; ++g) { bestD[g] = __builtin_inff(); bestI[g] = 0x7fffffff; }

  const int kt0 = ks * KT_PER;
  for (int kt = 0; kt < KT_PER; ++kt) {
    if ((kt & 7) == 0) __syncthreads();       // bound wave drift (keep B in WGP$)

    const int kbase = (kt0 + kt) * 16;
    const float* brow = cb + (size_t)(kbase + nsub) * DDIM + hi * 2;

    // prefetch next k-tile's codebook rows
    const int knext = (kt0 + ((kt + 1) & (KT_PER - 1))) * 16;
    __builtin_prefetch(cb + (size_t)(knext + nsub) * DDIM, 0, 1);

    v8f c0 = {0.f,0.f,0.f,0.f,0.f,0.f,0.f,0.f};
    v8f c1 = c0, c2 = c0, c3 = c0;            // 4 chains to hide WMMA latency
#pragma unroll 2
    for (int k0 = 0; k0 < DDIM; k0 += 16) {
      v2f a0 = *(const v2f*)(arow + k0);
      v2f b0 = *(const v2f*)(brow + k0);
      c0 = __builtin_amdgcn_wmma_f32_16x16x4_f32(false, a0, false, b0, (short)0, c0, false, false);
      v2f a1 = *(const v2f*)(arow + k0 + 4);
      v2f b1 = *(const v2f*)(brow + k0 + 4);
      c1 = __builtin_amdgcn_wmma_f32_16x16x4_f32(false, a1, false, b1, (short)0, c1, false, false);
      v2f a2 = *(const v2f*)(arow + k0 + 8);
      v2f b2 = *(const v2f*)(brow + k0 + 8);
      c2 = __builtin_amdgcn_wmma_f32_16x16x4_f32(false, a2, false, b2, (short)0, c2, false, false);
      v2f a3 = *(const v2f*)(arow + k0 + 12);
      v2f b3 = *(const v2f*)(brow + k0 + 12);
      c3 = __builtin_amdgcn_wmma_f32_16x16x4_f32(false, a3, false, b3, (short)0, c3, false, false);
    }

    const float e2n = e2[kbase + nsub];
    const int   idx = kbase + nsub;
#pragma unroll
    for (int g = 0; g < 8; ++g) {
      int m = g + hi * 8;                     // C/D layout: lane half -> M+8
      float cross = (c0[g] + c1[g]) + (c2[g] + c3[g]);
      float dist  = z2s[wave][m] - 2.0f * cross + e2n;
      if (dist < bestD[g] || (dist == bestD[g] && idx < bestI[g])) {
        bestD[g] = dist; bestI[g] = idx;
      }
    }
  }

  // ---- argmin reduce across the 16 lanes of each half ----
#pragma unroll
  for (int g = 0; g < 8; ++g) {
#pragma unroll
    for (int off = 8; off > 0; off >>= 1) {
      float od = __shfl_xor(bestD[g], off, 32);
      int   oi = __shfl_xor(bestI[g], off, 32);
      if (od < bestD[g] || (od == bestD[g] && oi < bestI[g])) {
        bestD[g] = od; bestI[g] = oi;
      }
    }
  }

  // lane 0 holds rows m=0..7 of this wave's subtile, lane 16 rows m=8..15
  const int rowbase = mblock * 128 + wave * 16;
  if (lane == 0) {
#pragma unroll
    for (int g = 0; g < 8; ++g) {
      partD[(rowbase + g) * KSPLIT + ks] = bestD[g];
      partI[(rowbase + g) * KSPLIT + ks] = bestI[g];
    }
  } else if (lane == 16) {
#pragma unroll
    for (int g = 0; g < 8; ++g) {
      partD[(rowbase + 8 + g) * KSPLIT + ks] = bestD[g];
      partI[(rowbase + 8 + g) * KSPLIT + ks] = bestI[g];
    }
  }
}

// ---------------------------------------------------------------------------
// Kernel 3: reduce K slices, gather codebook row, write both outputs.
// One wave per z row; bandwidth-bound (~134 MB total).
// ---------------------------------------------------------------------------
__global__ __launch_bounds__(256) void vq_finalize_kernel(const float* __restrict__ z_e,
                                                          const float* __restrict__ cb,
                                                          const float* __restrict__ partD,
                                                          const int*   __restrict__ partI,
                                                          float* __restrict__ out) {
  const int lane = threadIdx.x & 31;
  const int wave = threadIdx.x >> 5;
  const int row  = blockIdx.x * 8 + wave;           // 0..16383

  float bd = partD[row * KSPLIT];                   // ascending-k order: first-min
  int   bi = partI[row * KSPLIT];
#pragma unroll
  for (int s = 1; s < KSPLIT; ++s) {
    float od = partD[row * KSPLIT + s];
    int   oi = partI[row * KSPLIT + s];
    if (od < bd || (od == bd && oi < bi)) { bd = od; bi = oi; }
  }

  const float4* q4 = (const float4*)(cb + (size_t)bi * DDIM);
  const float4* z4 = (const float4*)(z_e + (size_t)row * DDIM);
  float4* o0 = (float4*)(out + (size_t)row * DDIM);                  // z_q_st
  float4* o1 = (float4*)(out + (size_t)(MROWS + row) * DDIM);        // z_q_bar
#pragma unroll
  for (int i = 0; i < 4; ++i) {
    int j = lane + 32 * i;
    float4 q = q4[j];
    float4 z = z4[j];
    float4 st;                                      // z_e + (z_q - z_e): same two
    st.x = z.x + (q.x - z.x);                       // rounding steps as reference
    st.y = z.y + (q.y - z.y);
    st.z = z.z + (q.z - z.z);
    st.w = z.w + (q.w - z.w);
    o0[j] = st;
    o1[j] = q;
  }
}

// ---------------------------------------------------------------------------
extern "C" void kernel_launch(void* const* d_in, const int* in_sizes, int n_in,
                              void* d_out, int out_size, void* d_ws, size_t ws_size,
                              hipStream_t stream) {
  const float* z_e = (const float*)d_in[0];   // [16,1024,512] f32
  const float* cb  = (const float*)d_in[1];   // [8192,512]    f32
  float* out   = (float*)d_out;               // [2,16,1024,512] f32
  float* wsf   = (float*)d_ws;
  float* e2    = wsf + WS_E2;                 // 8192 floats
  float* partD = wsf + WS_PARTD;              // 16384*4 floats
  int*   partI = (int*)(wsf + WS_PARTI);      // 16384*4 ints

  vq_e2_kernel<<<KCODES / 8, 256, 0, stream>>>(cb, e2);
  vq_main_kernel<<<dim3(MROWS / 128, KSPLIT), 256, 0, stream>>>(z_e, cb, e2, partD, partI);
  vq_finalize_kernel<<<MROWS / 8, 256, 0, stream>>>(z_e, cb, partD, partI, out);
}